// HAFT_6700148981829
// MI455X (gfx1250) — compile-verified
//
#include <hip/hip_runtime.h>
#include <math.h>

#define NPIX 16384   // 128*128
#define IMGN 2048    // B*C

// ---------------------------------------------------------------------------
// Shared-memory radix-2 FFT over 128 points, one thread per point.
// Caller loads bit-reversed data into sre/sim and __syncthreads() first.
// sign = -1 forward, +1 inverse (scaling applied by caller).
// ---------------------------------------------------------------------------
__device__ __forceinline__ void fft128_stage_loop(float* sre, float* sim, int i, float sign) {
  for (int len = 2; len <= 128; len <<= 1) {
    int half = len >> 1;
    int j = i & (len - 1);
    int jj = j & (half - 1);
    int p = (i - j) + jj;
    int q = p + half;
    float ang = sign * 6.28318530717958647692f * (float)jj / (float)len;
    float wi, wr;
    __sincosf(ang, &wi, &wr);
    float ur = sre[p], ui = sim[p];
    float vr = sre[q], vi = sim[q];
    float tr = vr * wr - vi * wi;
    float ti = vr * wi + vi * wr;
    float nr = (j < half) ? (ur + tr) : (ur - tr);
    float ni = (j < half) ? (ui + ti) : (ui - ti);
    __syncthreads();
    sre[i] = nr; sim[i] = ni;
    __syncthreads();
  }
}

// Pass 1: row FFTs of real input, write transposed complex.
__global__ void fft_rows_fwd1(const float* __restrict__ x, float2* __restrict__ cx) {
  __shared__ float sre[2][128], sim[2][128];
  int tid = threadIdx.x, r = tid >> 7, i = tid & 127;
  int nrow = blockIdx.x * 2 + r;
  int img = nrow >> 7, row = nrow & 127;
  int ib = __brev(i) >> 25;
  sre[r][i] = x[(size_t)img * NPIX + row * 128 + ib];
  sim[r][i] = 0.f;
  __syncthreads();
  fft128_stage_loop(sre[r], sim[r], i, -1.f);
  cx[(size_t)img * NPIX + i * 128 + row] = make_float2(sre[r][i], sim[r][i]);
}

// Pass 2: row FFTs of transposed data (= column FFT), emit mag/phase in original (u,v) layout.
__global__ void fft_rows_fwd2(const float2* __restrict__ cx, float* __restrict__ mag,
                              float* __restrict__ ph) {
  __shared__ float sre[2][128], sim[2][128];
  int tid = threadIdx.x, r = tid >> 7, i = tid & 127;
  int nrow = blockIdx.x * 2 + r;
  int img = nrow >> 7, row = nrow & 127;
  int ib = __brev(i) >> 25;
  float2 c = cx[(size_t)img * NPIX + row * 128 + ib];
  sre[r][i] = c.x; sim[r][i] = c.y;
  __syncthreads();
  fft128_stage_loop(sre[r], sim[r], i, -1.f);
  float rr = sre[r][i], ii = sim[r][i];
  size_t o = (size_t)img * NPIX + i * 128 + row;
  mag[o] = sqrtf(rr * rr + ii * ii);
  ph[o] = atan2f(ii, rr);
}

// Inverse pass 1: apply radial filters, iFFT along v, write transposed.
__global__ void ifft_rows_1(const float* __restrict__ mag, const float* __restrict__ ph,
                            const float* __restrict__ wfilt, float2* __restrict__ cx) {
  __shared__ float sre[2][128], sim[2][128];
  __shared__ float sw[2][32];
  int tid = threadIdx.x, r = tid >> 7, i = tid & 127;
  int nrow = blockIdx.x * 2 + r;
  int img = nrow >> 7, row = nrow & 127;
  if (i < 32) sw[r][i] = wfilt[(size_t)img * 32 + i];
  __syncthreads();
  int col = __brev(i) >> 25;
  float d = fmaxf(fabsf((float)row - 64.f), fabsf((float)col - 64.f));
  int ridx = (int)floorf(d * (15.f / 64.f));
  ridx = min(15, max(0, ridx));
  size_t src = (size_t)img * NPIX + row * 128 + col;
  float mg = mag[src] * sw[r][ridx];
  float p2 = ph[src] + sw[r][16 + ridx];
  float cc, ss;
  __sincosf(p2, &ss, &cc);
  sre[r][i] = mg * cc; sim[r][i] = mg * ss;
  __syncthreads();
  fft128_stage_loop(sre[r], sim[r], i, 1.f);
  cx[(size_t)img * NPIX + i * 128 + row] =
      make_float2(sre[r][i] * (1.f / 128.f), sim[r][i] * (1.f / 128.f));
}

// Inverse pass 2: iFFT along u, keep real part, back to original layout.
__global__ void ifft_rows_2(const float2* __restrict__ cx, float* __restrict__ enh) {
  __shared__ float sre[2][128], sim[2][128];
  int tid = threadIdx.x, r = tid >> 7, i = tid & 127;
  int nrow = blockIdx.x * 2 + r;
  int img = nrow >> 7, row = nrow & 127;
  int ib = __brev(i) >> 25;
  float2 c = cx[(size_t)img * NPIX + row * 128 + ib];
  sre[r][i] = c.x; sim[r][i] = c.y;
  __syncthreads();
  fft128_stage_loop(sre[r], sim[r], i, 1.f);
  enh[(size_t)img * NPIX + i * 128 + row] = sre[r][i] * (1.f / 128.f);
}

// ---------------------------------------------------------------------------
// Encoder stage 1: conv3x3 (1->16) + folded BN + ReLU + 32x32 avg pool -> (16,4,4)
// One block (256 threads) per image; 16 threads per pooled cell.
// ---------------------------------------------------------------------------
__global__ void enc1_kernel(const float* __restrict__ in, const float* __restrict__ w,
                            const float* __restrict__ cb, const float* __restrict__ g,
                            const float* __restrict__ be, const float* __restrict__ m,
                            const float* __restrict__ v, float* __restrict__ pooledOut) {
  __shared__ float sw[144];
  __shared__ float sscale[16], sshift[16];
  __shared__ float part[16][17];
  int tid = threadIdx.x;
  int img = blockIdx.x;
  if (tid < 144) sw[tid] = w[tid];
  if (tid < 16) {
    float sc = g[tid] * rsqrtf(v[tid] + 1e-5f);
    sscale[tid] = sc;
    sshift[tid] = cb[tid] * sc + be[tid] - m[tid] * sc;
  }
  __syncthreads();
  const float* im = in + (size_t)img * NPIX;
  int cell = tid >> 4, ln = tid & 15;
  int py = cell >> 2, px = cell & 3;
  float acc[16];
#pragma unroll
  for (int oc = 0; oc < 16; ++oc) acc[oc] = 0.f;
  for (int p = ln; p < 1024; p += 16) {
    int ly = p >> 5, lx = p & 31;
    int y = py * 32 + ly, x = px * 32 + lx;
    float nb[9];
#pragma unroll
    for (int dy = -1; dy <= 1; ++dy)
#pragma unroll
      for (int dx = -1; dx <= 1; ++dx) {
        int yy = y + dy, xx = x + dx;
        nb[(dy + 1) * 3 + dx + 1] =
            (yy >= 0 && yy < 128 && xx >= 0 && xx < 128) ? im[yy * 128 + xx] : 0.f;
      }
#pragma unroll
    for (int oc = 0; oc < 16; ++oc) {
      float s = 0.f;
#pragma unroll
      for (int k = 0; k < 9; ++k) s += sw[oc * 9 + k] * nb[k];
      acc[oc] += fmaxf(s * sscale[oc] + sshift[oc], 0.f);
    }
  }
  for (int oc = 0; oc < 16; ++oc) {
    part[cell][ln] = acc[oc];
    __syncthreads();
    if (ln == 0) {
      float s = 0.f;
      for (int k = 0; k < 16; ++k) s += part[cell][k];
      pooledOut[(size_t)img * 512 + oc * 16 + cell] = s * (1.f / 1024.f);
    }
    __syncthreads();
  }
}

// ---------------------------------------------------------------------------
// Encoder stage 2 + both MLPs, one block (128 threads) per image.
// ---------------------------------------------------------------------------
struct EncSide {
  const float *w2, *b2, *g2, *be2, *m2, *v2;  // conv2 + BN
  const float *wf, *bf;                       // FC 32->64
  const float *w1m, *b1m, *w2m, *b2m, *w3m, *b3m;  // MLP
};

__global__ void enc2_mlp_kernel(const float* __restrict__ pooled, EncSide S0, EncSide S1,
                                const float* __restrict__ lemb, float* __restrict__ wfilt) {
  __shared__ float sp[2][16][16];   // [enc][ic][pos]
  __shared__ float h2a[2][32][16];  // [enc][oc][pos]
  __shared__ float feat[2][32];
  __shared__ float cv[128];
  __shared__ float hb[256];
  __shared__ float hb2[256];
  __shared__ float logits[32];
  __shared__ float red[17];
  int tid = threadIdx.x;
  int img = blockIdx.x;
  const float* pbase = pooled + (size_t)img * 512;
  for (int l = tid; l < 512; l += 128) ((float*)sp)[l] = pbase[l];
  __syncthreads();
  // conv2 3x3 (16->32) on 4x4 SAME + BN + ReLU
  for (int s = 0; s < 8; ++s) {
    int o = tid + 128 * s;
    int enc = o >> 9, rem = o & 511, oc = rem >> 4, pos = rem & 15;
    const EncSide& S = enc ? S1 : S0;
    int py = pos >> 2, px = pos & 3;
    float acc = 0.f;
    for (int ic = 0; ic < 16; ++ic) {
      const float* wrow = S.w2 + ((oc * 16) + ic) * 9;
      for (int dy = -1; dy <= 1; ++dy) {
        int yy = py + dy;
        if (yy < 0 || yy > 3) continue;
        for (int dx = -1; dx <= 1; ++dx) {
          int xx = px + dx;
          if (xx < 0 || xx > 3) continue;
          acc += sp[enc][ic][yy * 4 + xx] * wrow[(dy + 1) * 3 + (dx + 1)];
        }
      }
    }
    float sc = S.g2[oc] * rsqrtf(S.v2[oc] + 1e-5f);
    float sh = S.b2[oc] * sc + S.be2[oc] - S.m2[oc] * sc;
    h2a[enc][oc][pos] = fmaxf(acc * sc + sh, 0.f);
  }
  __syncthreads();
  if (tid < 64) {
    int enc = tid >> 5, oc = tid & 31;
    float s = 0.f;
    for (int p2 = 0; p2 < 16; ++p2) s += h2a[enc][oc][p2];
    feat[enc][oc] = s * (1.f / 16.f);
  }
  __syncthreads();
  {  // FC 32->64 per side, concat, add level embedding
    int enc = tid >> 6, j = tid & 63;
    const EncSide& S = enc ? S1 : S0;
    float s = S.bf[j];
    for (int i2 = 0; i2 < 32; ++i2) s += feat[enc][i2] * S.wf[i2 * 64 + j];
    cv[tid] = s + lemb[tid];
  }
  __syncthreads();
  {  // MLP layer 1 (64->128) for both branches
    int j = tid;
    float sm = S0.b1m[j], sq = S1.b1m[j];
    for (int i2 = 0; i2 < 64; ++i2) {
      sm += cv[i2] * S0.w1m[i2 * 128 + j];
      sq += cv[64 + i2] * S1.w1m[i2 * 128 + j];
    }
    hb[j] = fmaxf(sm, 0.f);
    hb[128 + j] = fmaxf(sq, 0.f);
  }
  __syncthreads();
  {  // MLP layer 2 (128->128)
    int j = tid;
    float sm = S0.b2m[j], sq = S1.b2m[j];
    for (int i2 = 0; i2 < 128; ++i2) {
      sm += hb[i2] * S0.w2m[i2 * 128 + j];
      sq += hb[128 + i2] * S1.w2m[i2 * 128 + j];
    }
    hb2[j] = fmaxf(sm, 0.f);
    hb2[128 + j] = fmaxf(sq, 0.f);
  }
  __syncthreads();
  if (tid < 16) {  // mag logits
    float s = S0.b3m[tid];
    for (int i2 = 0; i2 < 128; ++i2) s += hb2[i2] * S0.w3m[i2 * 16 + tid];
    logits[tid] = s;
  } else if (tid < 32) {  // phase -> tanh
    int j = tid - 16;
    float s = S1.b3m[j];
    for (int i2 = 0; i2 < 128; ++i2) s += hb2[128 + i2] * S1.w3m[i2 * 16 + j];
    logits[tid] = tanhf(s);
  }
  __syncthreads();
  if (tid == 0) {
    float mx = logits[0];
    for (int k2 = 1; k2 < 16; ++k2) mx = fmaxf(mx, logits[k2]);
    float s = 0.f;
    for (int k2 = 0; k2 < 16; ++k2) {
      float e = expf(logits[k2] - mx);
      red[k2] = e;
      s += e;
    }
    red[16] = s;
  }
  __syncthreads();
  if (tid < 16) wfilt[(size_t)img * 32 + tid] = red[tid] / red[16];
  else if (tid < 32) wfilt[(size_t)img * 32 + tid] = logits[tid];
}

// ---------------------------------------------------------------------------
// 1x1 projection: out[b,o,hw] = sum_c proj_w[o,c] * enh[b,c,hw] + proj_b[o]
// fp32 WMMA 16x16x4. One block (8 waves) owns a full 256-row x 16-col output
// strip: each wave accumulates 2 M-tiles (m = wave*16 and wave*16+128), so the
// B panel (shared by all 16 M-tiles) is fetched from HBM exactly once and
// staged in LDS (double buffered, 64 K x 16 N per chunk).
// A fragment (16x4 fp32): lane<16 holds M=lane K={k,k+1}; lane>=16 K={k+2,k+3}.
// ---------------------------------------------------------------------------
typedef float v2f __attribute__((ext_vector_type(2)));
typedef float v8f __attribute__((ext_vector_type(8)));

__global__ void proj_wmma(const float* __restrict__ enh, const float* __restrict__ pw,
                          const float* __restrict__ pb, float* __restrict__ out) {
  __shared__ float Bs[2][64][16];  // [buf][k][n]  (4 KB per buffer)
  int tid = threadIdx.x;
  int wave = tid >> 5;
  int lane = tid & 31;
  int lo = lane & 15, hi = lane >> 4;
  int m0a = wave * 16;        // first M tile
  int m0b = wave * 16 + 128;  // second M tile
  int hw0 = blockIdx.x * 16;
  int bb = blockIdx.y;
  const float* eb = enh + (size_t)bb * 256 * NPIX;

  v8f acc0, acc1;
#pragma unroll
  for (int r2 = 0; r2 < 8; ++r2) {
    acc0[r2] = pb[m0a + r2 + 8 * hi];
    acc1[r2] = pb[m0b + r2 + 8 * hi];
  }

  int ldn = tid & 15, ldk = tid >> 4;  // B loader: 16 k-rows per pass, 4 passes
  // preload chunk 0
#pragma unroll
  for (int kk = 0; kk < 4; ++kk)
    Bs[0][ldk + kk * 16][ldn] = eb[(size_t)(ldk + kk * 16) * NPIX + hw0 + ldn];
  __syncthreads();

  for (int c = 0; c < 4; ++c) {
    int buf = c & 1;
    if (c + 1 < 4) {
      int k0n = (c + 1) * 64;
#pragma unroll
      for (int kk = 0; kk < 4; ++kk)
        Bs[buf ^ 1][ldk + kk * 16][ldn] = eb[(size_t)(k0n + ldk + kk * 16) * NPIX + hw0 + ldn];
    }
    int k0 = c * 64;
#pragma unroll 4
    for (int ks = 0; ks < 64; ks += 4) {
      int kl = ks + 2 * hi;       // lane-local K within chunk
      int ka = k0 + kl;           // global K
      v2f a0, a1, bv;
      a0[0] = pw[(m0a + lo) * 256 + ka];
      a0[1] = pw[(m0a + lo) * 256 + ka + 1];
      a1[0] = pw[(m0b + lo) * 256 + ka];
      a1[1] = pw[(m0b + lo) * 256 + ka + 1];
      bv[0] = Bs[buf][kl][lo];
      bv[1] = Bs[buf][kl + 1][lo];
      acc0 = __builtin_amdgcn_wmma_f32_16x16x4_f32(false, a0, false, bv, (short)0, acc0,
                                                   false, false);
      acc1 = __builtin_amdgcn_wmma_f32_16x16x4_f32(false, a1, false, bv, (short)0, acc1,
                                                   false, false);
    }
    __syncthreads();
  }

#pragma unroll
  for (int r2 = 0; r2 < 8; ++r2) {
    out[((size_t)bb * 256 + m0a + r2 + 8 * hi) * NPIX + hw0 + lo] = acc0[r2];
    out[((size_t)bb * 256 + m0b + r2 + 8 * hi) * NPIX + hw0 + lo] = acc1[r2];
  }
}

// ---------------------------------------------------------------------------
extern "C" void kernel_launch(void* const* d_in, const int* in_sizes, int n_in,
                              void* d_out, int out_size, void* d_ws, size_t ws_size,
                              hipStream_t stream) {
  (void)in_sizes; (void)n_in; (void)out_size; (void)ws_size;
  // Input order = setup_inputs() dict insertion order.
  const float* x = (const float*)d_in[0];
  const int EM = 1, EP = 15;  // enc_mag / enc_phase bases (w1,b1,g1,be1,m1,v1,w2,b2,g2,be2,m2,v2,wf,bf)
  const float* lemb = (const float*)d_in[29];
  const int MM = 30, PM = 36;  // mag_mlp / phase_mlp bases (w1,b1,w2,b2,w3,b3)
  const float* proj_w = (const float*)d_in[42];
  const float* proj_b = (const float*)d_in[43];

  // Workspace layout (floats)
  float* ws = (float*)d_ws;
  float2* cplx = (float2*)ws;                          // IMGN*NPIX complex
  float* mag = ws + (size_t)IMGN * NPIX * 2;           // IMGN*NPIX
  float* phase = mag + (size_t)IMGN * NPIX;            // IMGN*NPIX
  float* pooled = phase + (size_t)IMGN * NPIX;         // IMGN*512
  float* wfilt = pooled + (size_t)IMGN * 512;          // IMGN*32
  float* enh = wfilt + (size_t)IMGN * 32;              // IMGN*NPIX

  const int fftBlocks = IMGN * 128 / 2;

  fft_rows_fwd1<<<fftBlocks, 256, 0, stream>>>(x, cplx);
  fft_rows_fwd2<<<fftBlocks, 256, 0, stream>>>(cplx, mag, phase);

  enc1_kernel<<<IMGN, 256, 0, stream>>>(
      mag, (const float*)d_in[EM + 0], (const float*)d_in[EM + 1],
      (const float*)d_in[EM + 2], (const float*)d_in[EM + 3],
      (const float*)d_in[EM + 4], (const float*)d_in[EM + 5], pooled);
  enc1_kernel<<<IMGN, 256, 0, stream>>>(
      phase, (const float*)d_in[EP + 0], (const float*)d_in[EP + 1],
      (const float*)d_in[EP + 2], (const float*)d_in[EP + 3],
      (const float*)d_in[EP + 4], (const float*)d_in[EP + 5], pooled + 256);

  EncSide S0, S1;
  S0.w2 = (const float*)d_in[EM + 6];  S0.b2 = (const float*)d_in[EM + 7];
  S0.g2 = (const float*)d_in[EM + 8];  S0.be2 = (const float*)d_in[EM + 9];
  S0.m2 = (const float*)d_in[EM + 10]; S0.v2 = (const float*)d_in[EM + 11];
  S0.wf = (const float*)d_in[EM + 12]; S0.bf = (const float*)d_in[EM + 13];
  S0.w1m = (const float*)d_in[MM + 0]; S0.b1m = (const float*)d_in[MM + 1];
  S0.w2m = (const float*)d_in[MM + 2]; S0.b2m = (const float*)d_in[MM + 3];
  S0.w3m = (const float*)d_in[MM + 4]; S0.b3m = (const float*)d_in[MM + 5];
  S1.w2 = (const float*)d_in[EP + 6];  S1.b2 = (const float*)d_in[EP + 7];
  S1.g2 = (const float*)d_in[EP + 8];  S1.be2 = (const float*)d_in[EP + 9];
  S1.m2 = (const float*)d_in[EP + 10]; S1.v2 = (const float*)d_in[EP + 11];
  S1.wf = (const float*)d_in[EP + 12]; S1.bf = (const float*)d_in[EP + 13];
  S1.w1m = (const float*)d_in[PM + 0]; S1.b1m = (const float*)d_in[PM + 1];
  S1.w2m = (const float*)d_in[PM + 2]; S1.b2m = (const float*)d_in[PM + 3];
  S1.w3m = (const float*)d_in[PM + 4]; S1.b3m = (const float*)d_in[PM + 5];

  enc2_mlp_kernel<<<IMGN, 128, 0, stream>>>(pooled, S0, S1, lemb, wfilt);

  ifft_rows_1<<<fftBlocks, 256, 0, stream>>>(mag, phase, wfilt, cplx);
  ifft_rows_2<<<fftBlocks, 256, 0, stream>>>(cplx, enh);

  proj_wmma<<<dim3(1024, 8), 256, 0, stream>>>(enh, proj_w, proj_b, (float*)d_out);
}